// FilteredBackProjection_56573309223833
// MI455X (gfx1250) — compile-verified
//
#include <hip/hip_runtime.h>
#include <math.h>

typedef float v2f __attribute__((ext_vector_type(2)));
typedef float v8f __attribute__((ext_vector_type(8)));

#define H_    512
#define W_    512
#define NA    180
#define ND    729
#define BATCH 2
#define ROWS  (BATCH * NA)   // 360 sinogram rows
#define MP    384            // padded M (24 * 16)
#define KP    732            // padded K (183 * 4)
#define NP    736            // padded N (46 * 16)
#define MT    (MP / 16)      // 24
#define NT    (NP / 16)      // 46
#define PI_F  3.14159265358979323846f

// ---------------------------------------------------------------- init
__global__ void fbp_init(unsigned* maxcell) {
    if (threadIdx.x == 0 && blockIdx.x == 0) *maxcell = 0u;  // min of monotone map
}

// ----------------------------------------------- spatial filter c = Re(ifft(f))
// f[0]=0.25, f[odd k]=-1/(pi^2 k^2), else 0.  c[d] = (1/D) * sum_k f[k]*cos(2*pi*k*d/D)
// Use (k*d) mod D so the cos argument stays in [0, 2*pi) -> accurate fp32.
__global__ void fbp_filter_c(float* __restrict__ c) {
    int d = blockIdx.x * blockDim.x + threadIdx.x;
    if (d >= ND) return;
    float s = 0.25f;
    const float w = 2.0f * PI_F / (float)ND;
    for (int k = 1; k < ND; k += 2) {
        float fk = -1.0f / (PI_F * PI_F * (float)k * (float)k);
        int p = (k * d) % ND;
        s += fk * cosf(w * (float)p);
    }
    c[d] = s / (float)ND;
}

// ---------------------------------------------------------------- packing
__global__ void fbp_pack_spad(const float* __restrict__ sino, float* __restrict__ Spad) {
    int i = blockIdx.x * blockDim.x + threadIdx.x;
    if (i >= MP * KP) return;
    int r = i / KP, k = i % KP;
    Spad[i] = (r < ROWS && k < ND) ? sino[r * ND + k] : 0.0f;
}

__global__ void fbp_pack_bpad(const float* __restrict__ c, float* __restrict__ Bpad) {
    int i = blockIdx.x * blockDim.x + threadIdx.x;
    if (i >= KP * NP) return;
    int m = i / NP, n = i % NP;
    float v = 0.0f;
    if (m < ND && n < ND) {
        int d = n - m + ND;          // (n-m) in (-729, 729)
        if (d >= ND) d -= ND;
        v = c[d];
    }
    Bpad[i] = v;
}

// --------------------------------------------------- fp32 WMMA GEMM (filtering)
// One wave32 computes one 16x16 tile of filtered = Spad * Bpad via
// V_WMMA_F32_16X16X4_F32, K-loop over 732 in steps of 4.
// Fragment layouts per ISA 7.12.2 (32-bit A 16x4, B 4x16, C/D 16x16).
__global__ __launch_bounds__(256) void fbp_gemm_wmma(const float* __restrict__ Spad,
                                                     const float* __restrict__ Bpad,
                                                     float* __restrict__ Fpad) {
    int wave = blockIdx.x * 8 + (threadIdx.x >> 5);  // exact grid: 138*8 = 1104 = MT*NT
    int lane = threadIdx.x & 31;
    int half = lane >> 4;    // lanes 16-31 hold the upper K pair / upper M rows
    int l    = lane & 15;
    int tm   = wave / NT;
    int tn   = wave % NT;

    const float* arow = Spad + (size_t)(tm * 16 + l) * KP;
    const float* bcol = Bpad + tn * 16 + l;

    v8f acc = {};
    for (int k = 0; k < KP; k += 4) {
        int kk = k + 2 * half;
        // A 16x4: lane l (half=0) holds K=k..k+1, lane l+16 holds K=k+2..k+3 (contiguous b64 load)
        v2f a = *(const v2f*)(arow + kk);
        // B 4x16: vgpr0 rows k+2*half, vgpr1 rows k+2*half+1, col = l
        v2f b;
        b.x = bcol[(size_t)kk * NP];
        b.y = bcol[(size_t)(kk + 1) * NP];
        acc = __builtin_amdgcn_wmma_f32_16x16x4_f32(
                  /*neg_a=*/false, a, /*neg_b=*/false, b,
                  /*c_mod=*/(short)0, acc, /*reuse_a=*/false, /*reuse_b=*/false);
    }
    // C/D 16x16: VGPR r -> row (tm*16 + r + 8*half), col (tn*16 + l)
    float* drow = Fpad + (size_t)(tm * 16 + 8 * half) * NP + tn * 16 + l;
#pragma unroll
    for (int r = 0; r < 8; ++r) drow[(size_t)r * NP] = acc[r];
}

// ------------------------------------------------------------- backprojection
__global__ __launch_bounds__(256) void fbp_backproject(const float* __restrict__ Fpad,
                                                       float* __restrict__ out,
                                                       unsigned* __restrict__ maxcell) {
    __shared__ float cs[NA];
    __shared__ float sn[NA];
    __shared__ float red[256];
    int t = threadIdx.x;
    if (t < NA) {
        float ang = (float)t * (PI_F / (float)(NA - 1));  // linspace(0, pi, 180)
        cs[t] = cosf(ang);
        sn[t] = sinf(ang);
    }
    __syncthreads();

    int gid = blockIdx.x * 256 + t;                  // exact grid: 2048*256 pixels
    int b   = gid >> 18;                             // / (512*512)
    int rem = gid & 262143;
    int y   = rem >> 9;
    int x   = rem & 511;
    float xc = (float)x - (float)(W_ / 2);
    float yc = (float)y - (float)(H_ / 2);

    const float* Fb = Fpad + (size_t)b * NA * NP;
    float acc = 0.0f;
    for (int a = 0; a < NA; ++a) {
        float rot = xc * cs[a] + yc * sn[a];
        float tt  = rot / (2.0f * PI_F) * (float)ND; // same op order as reference
        int idx   = (int)tt;                         // trunc toward zero, like astype(int32)
        idx = idx < 0 ? 0 : (idx > ND - 1 ? ND - 1 : idx);
        acc += Fb[(size_t)a * NP + idx];
    }
    float val = acc * (PI_F / (float)NA);
    out[gid] = val;

    // block max reduction -> global atomicMax via monotone uint mapping
    red[t] = val;
    __syncthreads();
    for (int s = 128; s > 0; s >>= 1) {
        if (t < s) red[t] = fmaxf(red[t], red[t + s]);
        __syncthreads();
    }
    if (t == 0) {
        unsigned u = __float_as_uint(red[0]);
        u = (u & 0x80000000u) ? ~u : (u | 0x80000000u);
        atomicMax(maxcell, u);
    }
}

// ---------------------------------------------------------------------- clip
__global__ void fbp_clip(float* __restrict__ out, const unsigned* __restrict__ maxcell, int n) {
    int i = blockIdx.x * blockDim.x + threadIdx.x;
    if (i >= n) return;
    unsigned u = *maxcell;
    float M = (u & 0x80000000u) ? __uint_as_float(u & 0x7FFFFFFFu) : __uint_as_float(~u);
    float v = out[i];
    out[i] = fminf(fmaxf(v, 0.0f), M);   // jnp.clip(recon, 0, recon.max())
}

// -------------------------------------------------------------------- launch
extern "C" void kernel_launch(void* const* d_in, const int* in_sizes, int n_in,
                              void* d_out, int out_size, void* d_ws, size_t ws_size,
                              hipStream_t stream) {
    (void)in_sizes; (void)n_in; (void)out_size; (void)ws_size;
    const float* sino = (const float*)d_in[0];
    float* out = (float*)d_out;

    char* ws = (char*)d_ws;
    size_t off = 0;
    auto carve = [&](size_t bytes) -> char* {
        char* p = ws + off;
        off = (off + bytes + 255) & ~(size_t)255;
        return p;
    };
    unsigned* maxcell = (unsigned*)carve(sizeof(unsigned));
    float*    cbuf    = (float*)   carve((size_t)ND * sizeof(float));
    float*    Spad    = (float*)   carve((size_t)MP * KP * sizeof(float));
    float*    Bpad    = (float*)   carve((size_t)KP * NP * sizeof(float));
    float*    Fpad    = (float*)   carve((size_t)MP * NP * sizeof(float));

    hipLaunchKernelGGL(fbp_init,        dim3(1),                       dim3(1),   0, stream, maxcell);
    hipLaunchKernelGGL(fbp_filter_c,    dim3((ND + 255) / 256),        dim3(256), 0, stream, cbuf);
    hipLaunchKernelGGL(fbp_pack_spad,   dim3((MP * KP + 255) / 256),   dim3(256), 0, stream, sino, Spad);
    hipLaunchKernelGGL(fbp_pack_bpad,   dim3((KP * NP + 255) / 256),   dim3(256), 0, stream, cbuf, Bpad);
    hipLaunchKernelGGL(fbp_gemm_wmma,   dim3((MT * NT) / 8),           dim3(256), 0, stream, Spad, Bpad, Fpad);
    hipLaunchKernelGGL(fbp_backproject, dim3((BATCH * H_ * W_) / 256), dim3(256), 0, stream, Fpad, out, maxcell);
    hipLaunchKernelGGL(fbp_clip,        dim3((BATCH * H_ * W_ + 255) / 256), dim3(256), 0, stream,
                       out, maxcell, BATCH * H_ * W_);
}